// Layer_12936441495793
// MI455X (gfx1250) — compile-verified
//
#include <hip/hip_runtime.h>

typedef __attribute__((ext_vector_type(2))) float v2f;
typedef __attribute__((ext_vector_type(8))) float v8f;

#define NUQ   100000
#define NIQ   50000
#define DDIM  128
#define EEDG  300000
#define NNOD  (NUQ + NIQ)          // 150000, divisible by 16
#define LDSW  132                  // padded row stride (floats): conflict-free + 16B aligned

// ---------------------------------------------------------------------------
// Kernel 1: fused node projections with V_WMMA_F32_16X16X4_F32.
//   out[n]  = P1[n] = x[n] @ W1^T + b1         (self-loop contribution)
//   S[n]    = P1[n] + Q2[n]                    (Q2 = x[n] @ W2^T)
//   Qb2[n]  = Q2[n] + b2
// Block = 512 threads = 16 waves; each block handles 16 node rows.
// Waves 0..7 compute W1 col-tiles 0..7; waves 8..15 compute W2 col-tiles.
// ---------------------------------------------------------------------------
__global__ __launch_bounds__(512)
void proj_wmma_kernel(const float* __restrict__ xu, const float* __restrict__ xi,
                      const float* __restrict__ W1, const float* __restrict__ b1,
                      const float* __restrict__ W2, const float* __restrict__ b2,
                      float* __restrict__ out, float* __restrict__ S,
                      float* __restrict__ Qb2)
{
    __shared__ float xls[16 * LDSW];   // X tile, 16 rows x 128 cols (padded)
    __shared__ float qls[16 * LDSW];   // Q2 tile exchange buffer

    const int row0 = blockIdx.x * 16;
    const int tid  = threadIdx.x;

    // Cooperative load of the 16x128 X tile (2048 floats, 4 per thread).
    {
        const int idx = tid * 4;
        const int r   = idx >> 7;          // 0..15
        const int c   = idx & 127;
        const int row = row0 + r;
        const float* src = (row < NUQ) ? (xu + (size_t)row * DDIM + c)
                                       : (xi + (size_t)(row - NUQ) * DDIM + c);
        *(float4*)&xls[r * LDSW + c] = *(const float4*)src;
    }
    __syncthreads();

    const int lane = tid & 31;
    const int wv   = tid >> 5;
    const int isW2 = wv >> 3;              // 0: W1 waves, 1: W2 waves
    const int c0   = (wv & 7) * 16;        // output column tile base
    const int l15  = lane & 15;
    const int lh   = lane >> 4;            // lane half selects K pair

    const float* __restrict__ W    = isW2 ? W2 : W1;
    const float* __restrict__ Arow = &xls[l15 * LDSW + 2 * lh];               // X[row0+l15][k+2*lh]
    const float* __restrict__ Brow = W + (size_t)(c0 + l15) * DDIM + 2 * lh;  // W[c0+l15][k+2*lh]

    v8f acc = {};
    #pragma unroll
    for (int k = 0; k < DDIM; k += 4) {
        v2f a = *(const v2f*)(Arow + k);
        v2f b = *(const v2f*)(Brow + k);
        acc = __builtin_amdgcn_wmma_f32_16x16x4_f32(
            /*neg_a=*/false, a, /*neg_b=*/false, b,
            /*c_mod=*/(short)0, acc, /*reuse_a=*/false, /*reuse_b=*/false);
    }

    const int col = c0 + l15;              // global output column
    // C/D layout: VGPR v, lane l -> M = v + 8*(l>>4), N = l&15.
    if (isW2) {
        const float bb2 = b2[col];
        #pragma unroll
        for (int v = 0; v < 8; ++v) {
            const int m   = v + 8 * lh;
            const float q = acc[v];
            qls[m * LDSW + col] = q;
            Qb2[(size_t)(row0 + m) * DDIM + col] = q + bb2;
        }
    }
    __syncthreads();
    if (!isW2) {
        const float bb1 = b1[col];
        #pragma unroll
        for (int v = 0; v < 8; ++v) {
            const int m   = v + 8 * lh;
            const float p = acc[v] + bb1;                 // P1 element
            const float q = qls[m * LDSW + col];          // matching Q2 element
            out[(size_t)(row0 + m) * DDIM + col] = p;
            S  [(size_t)(row0 + m) * DDIM + col] = p + q;
        }
    }
}

// ---------------------------------------------------------------------------
// Kernel 2: edge messages. One wave per edge (32 lanes x float4 = 128 cols).
//   msg = norm * (S[src] + Qb2[dst]);  atomically add into out[dst].
// Edge ids [0, E) are user->item, [E, 2E) are item->user.
// ---------------------------------------------------------------------------
__global__ __launch_bounds__(256)
void edge_kernel(const float* __restrict__ S, const float* __restrict__ Qb2,
                 const int* __restrict__ src_ui, const int* __restrict__ dst_ui,
                 const float* __restrict__ norm_ui,
                 const int* __restrict__ src_iu, const int* __restrict__ dst_iu,
                 const float* __restrict__ norm_iu,
                 float* __restrict__ out)
{
    const int gw   = (int)((blockIdx.x * blockDim.x + threadIdx.x) >> 5);
    const int lane = threadIdx.x & 31;
    if (gw >= 2 * EEDG) return;

    size_t srow, drow;
    float  nrm;
    if (gw < EEDG) {                                   // user -> item
        const int e = gw;
        srow = (size_t)src_ui[e] * DDIM;               // user node
        drow = (size_t)(NUQ + dst_ui[e]) * DDIM;       // item node
        nrm  = norm_ui[e];
    } else {                                           // item -> user
        const int e = gw - EEDG;
        srow = (size_t)(NUQ + src_iu[e]) * DDIM;       // item node
        drow = (size_t)dst_iu[e] * DDIM;               // user node
        nrm  = norm_iu[e];
    }

    const int c = lane * 4;
    float4 sv = *(const float4*)(S   + srow + c);
    float4 qv = *(const float4*)(Qb2 + drow + c);
    float* o  = out + drow + c;
    unsafeAtomicAdd(o + 0, nrm * (sv.x + qv.x));
    unsafeAtomicAdd(o + 1, nrm * (sv.y + qv.y));
    unsafeAtomicAdd(o + 2, nrm * (sv.z + qv.z));
    unsafeAtomicAdd(o + 3, nrm * (sv.w + qv.w));
}

// ---------------------------------------------------------------------------
// Kernel 3: per-node leaky_relu + L2 normalize, one wave per row.
// ---------------------------------------------------------------------------
__global__ __launch_bounds__(256)
void finalize_kernel(float* __restrict__ out)
{
    const int gw   = (int)((blockIdx.x * blockDim.x + threadIdx.x) >> 5);
    const int lane = threadIdx.x & 31;
    if (gw >= NNOD) return;

    const size_t base = (size_t)gw * DDIM + lane * 4;
    float4 v = *(float4*)(out + base);
    v.x = v.x > 0.0f ? v.x : 0.2f * v.x;
    v.y = v.y > 0.0f ? v.y : 0.2f * v.y;
    v.z = v.z > 0.0f ? v.z : 0.2f * v.z;
    v.w = v.w > 0.0f ? v.w : 0.2f * v.w;

    float ss = v.x * v.x + v.y * v.y + v.z * v.z + v.w * v.w;
    #pragma unroll
    for (int off = 16; off > 0; off >>= 1)
        ss += __shfl_xor(ss, off, 32);

    const float scale = 1.0f / fmaxf(sqrtf(ss), 1e-12f);
    v.x *= scale; v.y *= scale; v.z *= scale; v.w *= scale;
    *(float4*)(out + base) = v;
}

extern "C" void kernel_launch(void* const* d_in, const int* in_sizes, int n_in,
                              void* d_out, int out_size, void* d_ws, size_t ws_size,
                              hipStream_t stream) {
    (void)in_sizes; (void)n_in; (void)out_size; (void)ws_size;

    const float* xu      = (const float*)d_in[0];
    const float* xi      = (const float*)d_in[1];
    const float* W1      = (const float*)d_in[2];
    const float* b1      = (const float*)d_in[3];
    const float* W2      = (const float*)d_in[4];
    const float* b2      = (const float*)d_in[5];
    const int*   src_ui  = (const int*)d_in[6];
    const int*   dst_ui  = (const int*)d_in[7];
    const float* norm_ui = (const float*)d_in[8];
    const int*   src_iu  = (const int*)d_in[9];
    const int*   dst_iu  = (const int*)d_in[10];
    const float* norm_iu = (const float*)d_in[11];

    float* out = (float*)d_out;                       // [NNOD, D]: users then items
    float* S   = (float*)d_ws;                        // [NNOD, D]
    float* Qb2 = S + (size_t)NNOD * DDIM;             // [NNOD, D]

    // 1) Fused WMMA projections (also initializes d_out with the self-loop P1).
    proj_wmma_kernel<<<NNOD / 16, 512, 0, stream>>>(xu, xi, W1, b1, W2, b2, out, S, Qb2);

    // 2) Edge gather/scale/scatter-add (2E edges, 8 waves per 256-thread block).
    edge_kernel<<<(2 * EEDG) / 8, 256, 0, stream>>>(S, Qb2, src_ui, dst_ui, norm_ui,
                                                    src_iu, dst_iu, norm_iu, out);

    // 3) leaky_relu + L2 normalize in place (8 rows per block).
    finalize_kernel<<<NNOD / 8, 256, 0, stream>>>(out);
}